// Encoder_77773267796720
// MI455X (gfx1250) — compile-verified
//
#include <hip/hip_runtime.h>
#include <hip/hip_bf16.h>

// ---------------------------------------------------------------------------
// GCN VAE encoder for MI455X (gfx1250, wave32).
//   mu, logstd = GCN(X, edges) with shared normalization.
// Strategy: fp32 WMMA for the (tiny) GEMMs, fused 128-wide scatter for the
// (dominant, L2-resident) edge aggregation. mu/logstd GEMM+scatter fused via
// W-concatenation.
// ---------------------------------------------------------------------------

typedef __attribute__((ext_vector_type(2))) float v2f;
typedef __attribute__((ext_vector_type(8))) float v8f;

#define NODES_F 128     // feature width of GEMM outputs (HID, and 64+64 concat)

__device__ __forceinline__ void atomic_add_f32(float* p, float v) {
    // Lowers to global_atomic_add_f32 (no CAS loop) on gfx1250.
    unsafeAtomicAdd(p, v);
}

// ---------------- generic fill ----------------
__global__ void fill_f32(float* __restrict__ p, float v, long long n) {
    long long i = (long long)blockIdx.x * blockDim.x + threadIdx.x;
    if (i < n) p[i] = v;
}

// ---------------- degree accumulation (deg pre-initialized to 1.0 = self loop)
__global__ void deg_accum(const int* __restrict__ eidx, float* __restrict__ deg,
                          int E) {
    int e = blockIdx.x * blockDim.x + threadIdx.x;
    if (e < E) {
        int dst = eidx[E + e];           // col = edge_index[1][e]
        atomic_add_f32(&deg[dst], 1.0f);
    }
}

// ---------------- dis = 1/sqrt(deg), in place ----------------
__global__ void rsqrt_inplace(float* __restrict__ d, int n) {
    int i = blockIdx.x * blockDim.x + threadIdx.x;
    if (i < n) d[i] = rsqrtf(d[i]);
}

// ---------------- pack Wcat = [W_mu | W_logstd]  (128 x 128) ----------------
__global__ void pack_w(const float* __restrict__ Wmu, const float* __restrict__ Wls,
                       float* __restrict__ Wcat) {
    int i = blockIdx.x * blockDim.x + threadIdx.x;   // 16384 threads
    int k = i >> 7, j = i & 127;
    Wcat[i] = (j < 64) ? Wmu[k * 64 + j] : Wls[k * 64 + (j - 64)];
}

// ---------------- WMMA GEMM: C[N,128] = A[N,128] * B[128,128] ---------------
// 256 threads = 8 waves. B staged in 64KB LDS. Each wave: one 16-row stripe,
// 8 column tiles (8 x v8f accumulators), K-loop of 32 steps of K=4.
// Fragment layout per CDNA5 ISA 7.12.2:
//   A 16x4 f32 : lanes 0-15 hold {K=0,K=1} of row M=lane; lanes 16-31 {K=2,K=3}
//   B 4x16 f32 : VGPR0/1 rows striped across lanes, K split by lane half
//   C/D 16x16  : VGPR r -> M=r (lanes 0-15) / M=r+8 (lanes 16-31), N=lane&15
__global__ __launch_bounds__(256) void gemm_n128_wmma(
    const float* __restrict__ A, const float* __restrict__ B,
    float* __restrict__ C, int nrows) {
    __shared__ float Bs[128 * 128];              // 64 KB of the 320 KB WGP LDS
    int t = threadIdx.x;
    #pragma unroll 4
    for (int i = t; i < 128 * 128; i += 256) Bs[i] = B[i];
    __syncthreads();

    int wave = t >> 5;
    int lane = t & 31;
    int tile = blockIdx.x * 8 + wave;            // 16-row tile index
    if (tile * 16 >= nrows) return;              // whole-wave exit: EXEC stays all-1s
    int row0 = tile * 16;
    int m  = lane & 15;                          // row / column within tile
    int kh = lane >> 4;                          // K half-select (0 -> K0,K1 ; 1 -> K2,K3)

    const float* arow = A + (long long)(row0 + m) * 128;

    v8f acc[8] = {};                             // 8 column tiles of 16x16 f32
    for (int kb = 0; kb < 128; kb += 4) {
        int k0 = kb + 2 * kh;
        v2f a;
        a.x = arow[k0 + 0];
        a.y = arow[k0 + 1];
        #pragma unroll
        for (int c = 0; c < 8; ++c) {
            int col = c * 16 + m;
            v2f b;
            b.x = Bs[(k0 + 0) * 128 + col];      // ds_load from staged W
            b.y = Bs[(k0 + 1) * 128 + col];
            acc[c] = __builtin_amdgcn_wmma_f32_16x16x4_f32(
                false, a, false, b, (short)0, acc[c], false, false);
        }
    }
    // Write back per C/D layout.
    #pragma unroll
    for (int c = 0; c < 8; ++c) {
        #pragma unroll
        for (int r = 0; r < 8; ++r) {
            int mrow = r + 8 * kh;
            int ncol = c * 16 + m;
            C[(long long)(row0 + mrow) * 128 + ncol] = acc[c][r];
        }
    }
}

// ---------------- edge scatter: AGG[dst] += H[src] * (dis[src]*dis[dst]) ----
// One wave per edge (incl. N self loops appended), lane l handles feats 4l..4l+3.
__global__ __launch_bounds__(256) void scatter_edges(
    const float* __restrict__ H, const int* __restrict__ eidx,
    const float* __restrict__ dis, float* __restrict__ AGG,
    int E, int N) {
    int e = blockIdx.x * 8 + (threadIdx.x >> 5);
    if (e >= E + N) return;
    int lane = threadIdx.x & 31;
    int src, dst; float w;
    if (e < E) {
        src = eidx[e];
        dst = eidx[E + e];
        w = dis[src] * dis[dst];
    } else {
        src = dst = e - E;                       // self loop
        float d = dis[src];
        w = d * d;
    }
    const float4* hp = (const float4*)(H + (long long)src * 128);
    float4 v = hp[lane];                         // global_load_b128
    float* ap = AGG + (long long)dst * 128 + lane * 4;
    atomic_add_f32(ap + 0, v.x * w);
    atomic_add_f32(ap + 1, v.y * w);
    atomic_add_f32(ap + 2, v.z * w);
    atomic_add_f32(ap + 3, v.w * w);
}

// ---------------- bias + relu, in place (n = N*128) ----------------
__global__ void bias_relu(float* __restrict__ buf, const float* __restrict__ b,
                          long long n) {
    long long i = (long long)blockIdx.x * blockDim.x + threadIdx.x;
    if (i < n) {
        float v = buf[i] + b[i & 127];
        buf[i] = v > 0.0f ? v : 0.0f;
    }
}

// ---------------- split AGG2[N,128] into mu / logstd with biases ------------
__global__ void finalize_out(const float* __restrict__ AGG2,
                             const float* __restrict__ bmu,
                             const float* __restrict__ bls,
                             float* __restrict__ out, int N) {
    int i = blockIdx.x * blockDim.x + threadIdx.x;   // N*64 threads
    if (i >= N * 64) return;
    int n = i >> 6, j = i & 63;
    const float* row = AGG2 + (long long)n * 128;
    out[i]                      = row[j]      + bmu[j];   // mu
    out[(long long)N * 64 + i]  = row[64 + j] + bls[j];   // logstd
}

// ---------------------------------------------------------------------------
extern "C" void kernel_launch(void* const* d_in, const int* in_sizes, int n_in,
                              void* d_out, int out_size, void* d_ws, size_t ws_size,
                              hipStream_t stream) {
    const float* x    = (const float*)d_in[0];   // [N,128]
    const int*   eidx = (const int*)d_in[1];     // [2,E] flat: row=e, col=E+e
    const float* W1   = (const float*)d_in[2];   // [128,128]
    const float* b1   = (const float*)d_in[3];   // [128]
    const float* Wmu  = (const float*)d_in[4];   // [128,64]
    const float* bmu  = (const float*)d_in[5];   // [64]
    const float* Wls  = (const float*)d_in[6];   // [128,64]
    const float* bls  = (const float*)d_in[7];   // [64]
    float* out = (float*)d_out;

    const int N = in_sizes[0] / 128;
    const int E = in_sizes[1] / 2;
    const long long NF = (long long)N * 128;

    // Workspace layout (floats): dis[N] | Wcat[16384] | bufA[N*128] | bufB[N*128]
    float* dis  = (float*)d_ws;
    float* Wcat = dis + N;
    float* bufA = Wcat + 128 * 128;
    float* bufB = bufA + NF;

    const int B256 = 256;
    dim3 blk(B256);
    int gN    = (N + B256 - 1) / B256;
    int gE    = (E + B256 - 1) / B256;
    long long gNFll = (NF + B256 - 1) / B256;
    int gNF   = (int)gNFll;
    int gEdge = (E + N + 7) / 8;                 // 8 waves (edges) per block
    int gGemm = ((N / 16) + 7) / 8;              // 8 row-tiles per block
    int gOut  = (N * 64 + B256 - 1) / B256;

    // 1) normalization: deg (init 1.0 for self loop) -> dis = rsqrt(deg)
    fill_f32<<<gN, blk, 0, stream>>>(dis, 1.0f, N);
    deg_accum<<<gE, blk, 0, stream>>>(eidx, dis, E);
    rsqrt_inplace<<<gN, blk, 0, stream>>>(dis, N);

    // 2) H1 = X @ W1   (bufA)
    gemm_n128_wmma<<<gGemm, blk, 0, stream>>>(x, W1, bufA, N);

    // 3) AGG1 = scatter(H1)  (bufB)
    fill_f32<<<gNF, blk, 0, stream>>>(bufB, 0.0f, NF);
    scatter_edges<<<gEdge, blk, 0, stream>>>(bufA, eidx, dis, bufB, E, N);

    // 4) H2 = relu(AGG1 + b1) in place (bufB)
    bias_relu<<<gNF, blk, 0, stream>>>(bufB, b1, NF);

    // 5) H3 = H2 @ [W_mu | W_logstd]  (bufA)
    pack_w<<<(128 * 128) / B256, blk, 0, stream>>>(Wmu, Wls, Wcat);
    gemm_n128_wmma<<<gGemm, blk, 0, stream>>>(bufB, Wcat, bufA, N);

    // 6) AGG2 = scatter(H3)  (bufB, reused)
    fill_f32<<<gNF, blk, 0, stream>>>(bufB, 0.0f, NF);
    scatter_edges<<<gEdge, blk, 0, stream>>>(bufA, eidx, dis, bufB, E, N);

    // 7) mu / logstd with biases
    finalize_out<<<gOut, blk, 0, stream>>>(bufB, bmu, bls, out, N);
}